// VectorQuantizer_42271068127825
// MI455X (gfx1250) — compile-verified
//
#include <hip/hip_runtime.h>
#include <hip/hip_bf16.h>

#define NUM_EMB   1024
#define DIM       64
#define N_ROWS    65536            // 64*32*32
#define N_BLOCKS  512              // N_ROWS / (8 waves * 16 rows)

typedef __attribute__((ext_vector_type(2))) float v2f;
typedef __attribute__((ext_vector_type(4))) float v4f;
typedef __attribute__((ext_vector_type(8))) float v8f;

// ---------------------------------------------------------------------------
// Kernel 0: precompute ||e||^2 for every codebook entry -> ws[0..1023]
// ---------------------------------------------------------------------------
__global__ void vq_norms_kernel(const float* __restrict__ cb,
                                float* __restrict__ e2) {
  int e = blockIdx.x * blockDim.x + threadIdx.x;
  if (e < NUM_EMB) {
    const v4f* row = (const v4f*)(cb + e * DIM);
    float s = 0.f;
#pragma unroll
    for (int i = 0; i < DIM / 4; ++i) {
      v4f q = row[i];
      s += q.x * q.x + q.y * q.y + q.z * q.z + q.w * q.w;
    }
    e2[e] = s;
  }
}

// ---------------------------------------------------------------------------
// Kernel 1: distances via V_WMMA_F32_16X16X4_F32, row argmin, gather, loss
// One wave = one 16-row tile of X. 8 waves / block -> 128 rows / block.
// ---------------------------------------------------------------------------
__global__ __launch_bounds__(256) void vq_main_kernel(
    const float* __restrict__ X, const float* __restrict__ CB,
    const float* __restrict__ e2, float* __restrict__ out,
    float* __restrict__ partial) {
  const int lane   = threadIdx.x & 31;
  const int laneLo = lane & 15;
  const int laneHi = lane >> 4;          // 0 or 1
  const int wave   = threadIdx.x >> 5;
  const int tile   = blockIdx.x * 8 + wave;
  const int r0     = tile * 16;

  // --- preload A fragments (16x64 tile of X, 16 chunks of K=4) ---
  // ISA 16x4 f32 A layout: lanes 0-15 hold K={0,1}, lanes 16-31 hold K={2,3}
  v2f Afrag[16];
  {
    const float* arow = X + (size_t)(r0 + laneLo) * DIM + laneHi * 2;
#pragma unroll
    for (int kc = 0; kc < 16; ++kc)
      Afrag[kc] = *(const v2f*)(arow + kc * 4);
  }

  // running per-row (min, argmin); VGPR j <-> row j (+8 for upper half-wave)
  float minv[8];
  int   mini[8];
#pragma unroll
  for (int j = 0; j < 8; ++j) { minv[j] = 3.4e38f; mini[j] = 0; }

  for (int nc = 0; nc < NUM_EMB / 16; ++nc) {
    const int n = nc * 16 + laneLo;                 // this lane's column
    const float* brow = CB + (size_t)n * DIM + laneHi * 2;
    v8f acc0 = {};
    v8f acc1 = {};
#pragma unroll
    for (int kc = 0; kc < 16; kc += 2) {
      v2f b0 = *(const v2f*)(brow + kc * 4);
      v2f b1 = *(const v2f*)(brow + kc * 4 + 4);
      acc0 = __builtin_amdgcn_wmma_f32_16x16x4_f32(
          false, Afrag[kc],     false, b0, (short)0, acc0, false, false);
      acc1 = __builtin_amdgcn_wmma_f32_16x16x4_f32(
          false, Afrag[kc + 1], false, b1, (short)0, acc1, false, false);
    }
    // dist' = ||e||^2 - 2 x.e   (||x||^2 dropped: constant per row)
    float en = e2[n];
#pragma unroll
    for (int j = 0; j < 8; ++j) {
      float d = en - 2.f * (acc0[j] + acc1[j]);
      if (d < minv[j]) { minv[j] = d; mini[j] = n; }   // keep first index on tie
    }
  }

  // --- butterfly argmin across the 16 lanes sharing each row set ---
#pragma unroll
  for (int m = 1; m < 16; m <<= 1) {
#pragma unroll
    for (int j = 0; j < 8; ++j) {
      float ov = __shfl_xor(minv[j], m, 32);
      int   oi = __shfl_xor(mini[j], m, 32);
      if (ov < minv[j] || (ov == minv[j] && oi < mini[j])) {
        minv[j] = ov; mini[j] = oi;
      }
    }
  }

  // --- gather codebook rows, write quantized output, accumulate loss ---
  float ssum = 0.f;
#pragma unroll
  for (int j = 0; j < 8; ++j) {
    int idx = mini[j];                         // uniform within each half-wave
    int m   = r0 + laneHi * 8 + j;
    int c   = laneLo * 4;
    v4f q = *(const v4f*)(CB + (size_t)idx * DIM + c);
    v4f x = *(const v4f*)(X + (size_t)m * DIM + c);
    *(v4f*)(out + (size_t)m * DIM + c) = q;
    v4f dq = q - x;
    ssum += dq.x * dq.x + dq.y * dq.y + dq.z * dq.z + dq.w * dq.w;
  }

  // wave reduce
#pragma unroll
  for (int m = 1; m < 32; m <<= 1) ssum += __shfl_xor(ssum, m, 32);

  __shared__ float wsum[8];
  if (lane == 0) wsum[wave] = ssum;
  __syncthreads();
  if (threadIdx.x == 0) {
    float t = 0.f;
#pragma unroll
    for (int w = 0; w < 8; ++w) t += wsum[w];   // fixed order -> deterministic
    partial[blockIdx.x] = t;
  }
}

// ---------------------------------------------------------------------------
// Kernel 2: deterministic reduction of per-block partials -> vq_loss
// ---------------------------------------------------------------------------
__global__ void vq_loss_kernel(const float* __restrict__ partial,
                               float* __restrict__ lossOut) {
  int lane = threadIdx.x;                      // 32 threads, 1 wave
  float s = 0.f;
  for (int i = lane; i < N_BLOCKS; i += 32) s += partial[i];
#pragma unroll
  for (int m = 1; m < 32; m <<= 1) s += __shfl_xor(s, m, 32);
  if (lane == 0)
    lossOut[0] = 1.25f * s / (float)((size_t)N_ROWS * DIM);
}

extern "C" void kernel_launch(void* const* d_in, const int* in_sizes, int n_in,
                              void* d_out, int out_size, void* d_ws, size_t ws_size,
                              hipStream_t stream) {
  const float* X  = (const float*)d_in[0];     // [65536, 64]
  const float* CB = (const float*)d_in[1];     // [1024, 64]
  float* out      = (float*)d_out;             // quantized_st, then loss scalar
  float* ws_f     = (float*)d_ws;
  float* e2       = ws_f;                      // 1024 floats
  float* partial  = ws_f + NUM_EMB;            // 512 floats
  float* loss     = out + (out_size - 1);      // last element = vq_loss

  vq_norms_kernel<<<NUM_EMB / 256, 256, 0, stream>>>(CB, e2);
  vq_main_kernel<<<N_BLOCKS, 256, 0, stream>>>(X, CB, e2, out, partial);
  vq_loss_kernel<<<1, 32, 0, stream>>>(partial, loss);
}